// AttentionContext_85341000172033
// MI455X (gfx1250) — compile-verified
//
#include <hip/hip_runtime.h>
#include <math.h>

typedef __attribute__((ext_vector_type(2))) float v2f;
typedef __attribute__((ext_vector_type(4))) float v4f;
typedef __attribute__((ext_vector_type(8))) float v8f;

#define BB 64
#define TT 2048
#define DD 512

// ---------------------------------------------------------------------------
// Small f32 WMMA GEMM: C[m,n] = sum_k A[m,k] * (BT ? Bm[n,k] : Bm[k,n]) + bias[n]
// One 16x16 output tile per wave, V_WMMA_F32_16X16X4_F32 along K.
// Lane mapping (ISA 7.12.2, 32-bit operands, wave32):
//   A 16x4 : lane%16 = m, a[v] = A[m][k0 + 2*(lane/16) + v]
//   B 4x16 : lane%16 = n, b[v] = B[k0 + 2*(lane/16) + v][n]
//   C 16x16: lane%16 = n, c[v] = C[v + 8*(lane/16)][n]
// Guards are wave-uniform so EXEC stays all-ones around the WMMA.
// ---------------------------------------------------------------------------
template <bool BT>
__global__ __launch_bounds__(256) void gemm_wmma_f32(
    const float* __restrict__ A, const float* __restrict__ Bm,
    const float* __restrict__ bias, float* __restrict__ C,
    int M, int N, int K)
{
  const int wave = threadIdx.x >> 5;
  const int lane = threadIdx.x & 31;
  const int half = lane >> 4;
  const int lm   = lane & 15;

  const int ntiles = N >> 4;
  const int tile   = blockIdx.x * 8 + wave;
  const int mt     = tile / ntiles;
  const int nt     = tile - mt * ntiles;
  if (mt * 16 >= M) return;  // wave-uniform

  const float* arow = A + (size_t)(mt * 16 + lm) * K;

  v8f acc = {};
  for (int k0 = 0; k0 < K; k0 += 4) {
    const int k = k0 + 2 * half;
    v2f a;
    a.x = arow[k];
    a.y = arow[k + 1];
    v2f b;
    if (BT) {
      const float* brow = Bm + (size_t)(nt * 16 + lm) * K;
      b.x = brow[k];
      b.y = brow[k + 1];
    } else {
      const float* bcol = Bm + (nt * 16 + lm);
      b.x = bcol[(size_t)k * N];
      b.y = bcol[(size_t)(k + 1) * N];
    }
    acc = __builtin_amdgcn_wmma_f32_16x16x4_f32(
        /*neg_a=*/false, a, /*neg_b=*/false, b,
        /*c_mod=*/(short)0, acc, /*reuse_a=*/false, /*reuse_b=*/false);
  }

  const int n  = nt * 16 + lm;
  const float bv = bias ? bias[n] : 0.0f;
#pragma unroll
  for (int v = 0; v < 8; ++v) {
    const int m = mt * 16 + v + 8 * half;
    C[(size_t)m * N + n] = acc[v] + bv;
  }
}

// ---------------------------------------------------------------------------
// Single streaming pass over h (256 MB, read-once -> nontemporal):
//   e[b,t] = dot(u[b,:], h[b,t,:]),  r[b,t] = sum_k h[b,t,k]
// One wave per (b,t) row of 512 floats; b128 loads; wave32 shfl reduction.
// ---------------------------------------------------------------------------
__global__ __launch_bounds__(256) void row_pass(
    const float* __restrict__ h, const float* __restrict__ u,
    float* __restrict__ e, float* __restrict__ r)
{
  const int wave = threadIdx.x >> 5;
  const int lane = threadIdx.x & 31;
  const int row  = blockIdx.x * 8 + wave;   // row in [0, B*T)
  const int b    = row >> 11;               // T = 2048

  const v4f* hp = (const v4f*)(h + (size_t)row * DD);
  const v4f* up = (const v4f*)(u + (size_t)b * DD);

  float dot = 0.0f, sum = 0.0f;
#pragma unroll
  for (int i = 0; i < 4; ++i) {
    const v4f hv = __builtin_nontemporal_load(&hp[lane + 32 * i]);
    const v4f uv = up[lane + 32 * i];
    dot += hv.x * uv.x + hv.y * uv.y + hv.z * uv.z + hv.w * uv.w;
    sum += hv.x + hv.y + hv.z + hv.w;
  }
#pragma unroll
  for (int off = 16; off > 0; off >>= 1) {
    dot += __shfl_xor(dot, off, 32);
    sum += __shfl_xor(sum, off, 32);
  }
  if (lane == 0) {
    e[row] = dot;
    r[row] = sum;
  }
}

// ---------------------------------------------------------------------------
// Per-batch softmax over T and context: out[b,t] = softmax_t(e)[b,t] * r[b,t]
// One block per b, 256 threads, 8 elements/thread, LDS block reductions.
// ---------------------------------------------------------------------------
__global__ __launch_bounds__(256) void softmax_ctx(
    const float* __restrict__ e, const float* __restrict__ r,
    float* __restrict__ out)
{
  __shared__ float red[256];
  const int b   = blockIdx.x;
  const int tid = threadIdx.x;
  const float* eb = e + (size_t)b * TT;
  const float* rb = r + (size_t)b * TT;

  float vals[8];
  float m = -INFINITY;
#pragma unroll
  for (int i = 0; i < 8; ++i) {
    vals[i] = eb[tid + 256 * i];
    m = fmaxf(m, vals[i]);
  }
  red[tid] = m;
  __syncthreads();
  for (int s = 128; s > 0; s >>= 1) {
    if (tid < s) red[tid] = fmaxf(red[tid], red[tid + s]);
    __syncthreads();
  }
  m = red[0];
  __syncthreads();

  float sum = 0.0f;
#pragma unroll
  for (int i = 0; i < 8; ++i) {
    vals[i] = __expf(vals[i] - m);
    sum += vals[i];
  }
  red[tid] = sum;
  __syncthreads();
  for (int s = 128; s > 0; s >>= 1) {
    if (tid < s) red[tid] += red[tid + s];
    __syncthreads();
  }
  const float inv = 1.0f / red[0];

#pragma unroll
  for (int i = 0; i < 8; ++i) {
    out[(size_t)b * TT + tid + 256 * i] = vals[i] * inv * rb[tid + 256 * i];
  }
}

// ---------------------------------------------------------------------------
// Orchestration. psi_b (d_in[5]) provably drops out: it shifts e[b,*] by a
// per-b constant and softmax over t is shift-invariant.
// Workspace layout (floats): phi_s[64*512] | u[64*512] | e[64*2048] | r[64*2048]
// = 1.25 MB total.
// ---------------------------------------------------------------------------
extern "C" void kernel_launch(void* const* d_in, const int* in_sizes, int n_in,
                              void* d_out, int out_size, void* d_ws, size_t ws_size,
                              hipStream_t stream) {
  (void)in_sizes; (void)n_in; (void)out_size; (void)ws_size;

  const float* s     = (const float*)d_in[0];
  const float* h     = (const float*)d_in[1];
  const float* phi_w = (const float*)d_in[2];
  const float* phi_b = (const float*)d_in[3];
  const float* psi_w = (const float*)d_in[4];

  float* ws    = (float*)d_ws;
  float* phi_s = ws;                 // [64, 512]
  float* u     = phi_s + BB * DD;    // [64, 512]
  float* e     = u + BB * DD;        // [64, 2048]
  float* r     = e + BB * TT;        // [64, 2048]

  // phi_s[b,a] = s[b,:] . phi_w[a,:] + phi_b[a]   (B accessed transposed)
  gemm_wmma_f32<true><<<16, 256, 0, stream>>>(s, phi_w, phi_b, phi_s, BB, DD, DD);
  // u[b,k] = phi_s[b,:] . psi_w[:,k]              (B row-major)
  gemm_wmma_f32<false><<<16, 256, 0, stream>>>(phi_s, psi_w, nullptr, u, BB, DD, DD);
  // Single HBM-bound pass over h: scores + row sums.
  row_pass<<<(BB * TT) / 8, 256, 0, stream>>>(h, u, e, r);
  // Softmax over T and context output.
  softmax_ctx<<<BB, 256, 0, stream>>>(e, r, (float*)d_out);
}